// GCN_6330781794593
// MI455X (gfx1250) — compile-verified
//
#include <hip/hip_runtime.h>
#include <math.h>

#define D_IN   128
#define D_HID  128
#define D_OUT  40
#define T2S    48   // padded stride for layer-2 pre-scatter buffer

typedef float v2f __attribute__((ext_vector_type(2)));
typedef float v8f __attribute__((ext_vector_type(8)));

__device__ __forceinline__ v8f wmma_f32_16x16x4(v2f a, v2f b, v8f c) {
    // 8 args: (neg_a, A, neg_b, B, c_mod, C, reuse_a, reuse_b)
    return __builtin_amdgcn_wmma_f32_16x16x4_f32(false, a, false, b, (short)0, c, false, false);
}

// CDNA5 async global->LDS copy (ASYNCcnt tracked), 16 bytes per call.
// VDST = LDS byte offset in a VGPR, VADDR = 64-bit global address.
__device__ __forceinline__ void async_g2l_b128(unsigned lds_off, const float* gaddr) {
    asm volatile("global_load_async_to_lds_b128 %0, %1, off"
                 :: "v"(lds_off), "v"((unsigned long long)(uintptr_t)gaddr)
                 : "memory");
}
__device__ __forceinline__ void wait_async0() {
    asm volatile("s_wait_asynccnt 0x0" ::: "memory");
}

// ---------------- utility kernels ----------------
__global__ void k_zero_f32(float* __restrict__ p, int n) {
    int i = blockIdx.x * blockDim.x + threadIdx.x;
    if (i < n) p[i] = 0.0f;
}

__global__ void k_init_deg(float* __restrict__ deg, int n) {
    int i = blockIdx.x * blockDim.x + threadIdx.x;
    if (i < n) deg[i] = 1.0f;   // self loop
}

__global__ void k_deg(const long long* __restrict__ ei, float* __restrict__ deg, int E) {
    int e = blockIdx.x * blockDim.x + threadIdx.x;
    if (e < E) atomicAdd(&deg[(int)ei[(size_t)E + e]], 1.0f);
}

__global__ void k_rsqrt_inplace(float* __restrict__ d, int n) {
    int i = blockIdx.x * blockDim.x + threadIdx.x;
    if (i < n) d[i] = rsqrtf(d[i]);   // deg >= 1 always (self loops)
}

// ---------------- GEMM1: T1 = x @ W1   [N,128]x[128,128] ----------------
// block = 256 threads = 8 waves; each wave owns one 16-wide N tile and 4 M tiles
// (block covers 64 rows). W1 half (32KB) and x tile (16KB) staged in LDS via
// async global->LDS; B fragments preloaded to registers per K-half so the hot
// loop is ds_load(A) + v_wmma only.
__global__ void k_gemm1(const float* __restrict__ x, const float* __restrict__ W1,
                        float* __restrict__ T1, int N) {
    __shared__ float lW[64 * 128];   // W1 rows kb..kb+63, [k][n]
    __shared__ float lX[64 * 64];    // x rows mB..mB+63, cols kb..kb+63, [m][k]
    const int lane = threadIdx.x & 31;
    const int wave = threadIdx.x >> 5;
    const int n0   = wave * 16;
    const int mB   = blockIdx.x * 64;
    const int half = lane >> 4;      // 0: K pair {0,1}, 1: K pair {2,3}
    const int l16  = lane & 15;

    v8f acc[4] = {};

    for (int kb = 0; kb < 128; kb += 64) {
        // stage W1 half: 8192 floats contiguous = 2048 x b128
        for (int idx = threadIdx.x; idx < 2048; idx += blockDim.x)
            async_g2l_b128((unsigned)(uintptr_t)&lW[idx * 4],
                           W1 + (size_t)kb * 128 + idx * 4);
        // stage x tile: 64 rows x 16 b128 segments
        for (int idx = threadIdx.x; idx < 1024; idx += blockDim.x) {
            int r = idx >> 4, sg = idx & 15;
            int m = mB + r; if (m >= N) m = N - 1;           // clamp; store guarded
            async_g2l_b128((unsigned)(uintptr_t)&lX[r * 64 + sg * 4],
                           x + (size_t)m * 128 + kb + sg * 4);
        }
        wait_async0();
        __syncthreads();

        // preload B fragments for this K-half into registers (16 steps x v2f)
        v2f bf[16];
        #pragma unroll
        for (int s = 0; s < 16; ++s) {
            const int kk = s * 4 + 2 * half;
            bf[s].x = lW[kk * 128 + n0 + l16];
            bf[s].y = lW[(kk + 1) * 128 + n0 + l16];
        }

        #pragma unroll
        for (int s = 0; s < 16; ++s) {
            const int kk = s * 4 + 2 * half;
            #pragma unroll
            for (int t = 0; t < 4; ++t) {
                v2f a = *(const v2f*)&lX[(t * 16 + l16) * 64 + kk];
                acc[t] = wmma_f32_16x16x4(a, bf[s], acc[t]);
            }
        }
        __syncthreads();
    }

    #pragma unroll
    for (int t = 0; t < 4; ++t)
        #pragma unroll
        for (int r = 0; r < 8; ++r) {
            int m = mB + t * 16 + r + half * 8;
            if (m < N) T1[(size_t)m * 128 + n0 + l16] = acc[t][r];
        }
}

// ---------------- GEMM2: T2 = H1 @ W2   [N,128]x[128,40] (padded to 48) -----
// block = 256 threads = 8 waves; each wave owns one 16-row M tile, all 3 N tiles.
__global__ void k_gemm2(const float* __restrict__ H1, const float* __restrict__ W2,
                        float* __restrict__ T2, int N) {
    __shared__ float lW[128 * T2S];
    const int lane = threadIdx.x & 31;
    const int wave = threadIdx.x >> 5;
    const int half = lane >> 4;
    const int l16  = lane & 15;

    for (int idx = threadIdx.x; idx < 128 * T2S; idx += blockDim.x) {
        int r = idx / T2S, c = idx % T2S;
        lW[idx] = (c < D_OUT) ? W2[(size_t)r * D_OUT + c] : 0.0f;
    }
    __syncthreads();

    const int m0 = blockIdx.x * 128 + wave * 16;
    int m = m0 + l16;
    if (m >= N) m = N - 1;
    const float* arow = H1 + (size_t)m * 128;

    v8f acc[3] = {};
    #pragma unroll
    for (int k = 0; k < 128; k += 4) {
        const int kk = k + 2 * half;
        v2f a = *(const v2f*)(arow + kk);
        #pragma unroll
        for (int nt = 0; nt < 3; ++nt) {
            v2f b;
            b.x = lW[kk * T2S + nt * 16 + l16];
            b.y = lW[(kk + 1) * T2S + nt * 16 + l16];
            acc[nt] = wmma_f32_16x16x4(a, b, acc[nt]);
        }
    }

    #pragma unroll
    for (int nt = 0; nt < 3; ++nt)
        #pragma unroll
        for (int r = 0; r < 8; ++r) {
            int mm = m0 + r + half * 8;
            if (mm < N) T2[(size_t)mm * T2S + nt * 16 + l16] = acc[nt][r];
        }
}

// ---------------- edge scatter, layer 1 (128-wide) ----------------
// one wave per edge; each lane moves float4 (32 lanes * 4 = 128)
__global__ void k_scatter1(const long long* __restrict__ ei, const float* __restrict__ dinv,
                           const float* __restrict__ T1, float* __restrict__ H1acc, int E) {
    const int lane = threadIdx.x & 31;
    const int e = blockIdx.x * (blockDim.x >> 5) + (threadIdx.x >> 5);
    if (e >= E) return;
    const int r = (int)ei[e];
    const int c = (int)ei[(size_t)E + e];
    const float w = dinv[r] * dinv[c];
    const float4 v = *(const float4*)(T1 + (size_t)r * 128 + lane * 4);
    float* dst = H1acc + (size_t)c * 128 + lane * 4;
    atomicAdd(dst + 0, v.x * w);
    atomicAdd(dst + 1, v.y * w);
    atomicAdd(dst + 2, v.z * w);
    atomicAdd(dst + 3, v.w * w);
}

// ---------------- epilogue 1: self-loop + bias + ReLU (in place) ----------------
__global__ void k_epi1(float* __restrict__ H1acc, const float* __restrict__ T1,
                       const float* __restrict__ dinv, const float* __restrict__ b1, int N) {
    int i = blockIdx.x * blockDim.x + threadIdx.x;
    if (i >= N * 128) return;
    int node = i >> 7, d = i & 127;
    float s = dinv[node]; s *= s;
    float v = H1acc[i] + T1[i] * s + b1[d];
    H1acc[i] = v > 0.0f ? v : 0.0f;
}

// ---------------- edge scatter, layer 2 (40-wide) ----------------
__global__ void k_scatter2(const long long* __restrict__ ei, const float* __restrict__ dinv,
                           const float* __restrict__ T2, float* __restrict__ H2acc, int E) {
    const int lane = threadIdx.x & 31;
    const int e = blockIdx.x * (blockDim.x >> 5) + (threadIdx.x >> 5);
    if (e >= E) return;
    const int r = (int)ei[e];
    const int c = (int)ei[(size_t)E + e];
    const float w = dinv[r] * dinv[c];
    const float* src = T2 + (size_t)r * T2S;
    float* dst = H2acc + (size_t)c * D_OUT;
    atomicAdd(dst + lane, src[lane] * w);
    if (lane < 8) atomicAdd(dst + 32 + lane, src[32 + lane] * w);
}

// ---------------- final: self-loop + bias + log_softmax (wave per node) ----------------
__global__ void k_final(const float* __restrict__ H2acc, const float* __restrict__ T2,
                        const float* __restrict__ dinv, const float* __restrict__ b2,
                        float* __restrict__ out, int N) {
    const int lane = threadIdx.x & 31;
    const int i = blockIdx.x * (blockDim.x >> 5) + (threadIdx.x >> 5);
    if (i >= N) return;
    float s = dinv[i]; s *= s;
    float v0 = H2acc[(size_t)i * D_OUT + lane] + T2[(size_t)i * T2S + lane] * s + b2[lane];
    float v1 = -__builtin_inff();
    if (lane < 8)
        v1 = H2acc[(size_t)i * D_OUT + 32 + lane] + T2[(size_t)i * T2S + 32 + lane] * s + b2[32 + lane];
    float m = fmaxf(v0, v1);
    #pragma unroll
    for (int o = 16; o > 0; o >>= 1) m = fmaxf(m, __shfl_xor(m, o, 32));
    float sum = expf(v0 - m) + (lane < 8 ? expf(v1 - m) : 0.0f);
    #pragma unroll
    for (int o = 16; o > 0; o >>= 1) sum += __shfl_xor(sum, o, 32);
    float L = m + logf(sum);
    out[(size_t)i * D_OUT + lane] = v0 - L;
    if (lane < 8) out[(size_t)i * D_OUT + 32 + lane] = v1 - L;
}

// ---------------- launcher ----------------
extern "C" void kernel_launch(void* const* d_in, const int* in_sizes, int n_in,
                              void* d_out, int out_size, void* d_ws, size_t ws_size,
                              hipStream_t stream) {
    const float*     x  = (const float*)d_in[0];
    const long long* ei = (const long long*)d_in[1];   // int64 edge_index [2,E]
    const float*     W1 = (const float*)d_in[2];
    const float*     b1 = (const float*)d_in[3];
    const float*     W2 = (const float*)d_in[4];
    const float*     b2 = (const float*)d_in[5];

    const int N = in_sizes[0] / D_IN;
    const int E = in_sizes[1] / 2;

    // workspace carve-out
    char*  ws  = (char*)d_ws;
    size_t off = 0;
    auto carve = [&](size_t bytes) -> void* {
        void* p = ws + off;
        off += (bytes + 255) & ~(size_t)255;
        return p;
    };
    float* dinv  = (float*)carve((size_t)N * 4);            // deg, then deg^-1/2 in place
    float* T1    = (float*)carve((size_t)N * D_HID * 4);    // x @ W1
    float* H1    = (float*)carve((size_t)N * D_HID * 4);    // scatter acc -> relu output
    float* T2    = (float*)carve((size_t)N * T2S * 4);      // H1 @ W2 (padded)
    float* H2    = (float*)carve((size_t)N * D_OUT * 4);    // scatter acc
    (void)ws_size; (void)n_in; (void)out_size;

    const int TB = 256;

    // degree + normalization
    k_init_deg<<<(N + TB - 1) / TB, TB, 0, stream>>>(dinv, N);
    k_deg<<<(E + TB - 1) / TB, TB, 0, stream>>>(ei, dinv, E);
    k_rsqrt_inplace<<<(N + TB - 1) / TB, TB, 0, stream>>>(dinv, N);

    // zero accumulators (poisoned workspace)
    k_zero_f32<<<((N * D_HID) + TB - 1) / TB, TB, 0, stream>>>(H1, N * D_HID);
    k_zero_f32<<<((N * D_OUT) + TB - 1) / TB, TB, 0, stream>>>(H2, N * D_OUT);

    // layer 1
    k_gemm1<<<(N + 63) / 64, TB, 0, stream>>>(x, W1, T1, N);
    k_scatter1<<<(E + 7) / 8, TB, 0, stream>>>(ei, dinv, T1, H1, E);
    k_epi1<<<((N * D_HID) + TB - 1) / TB, TB, 0, stream>>>(H1, T1, dinv, b1, N);

    // layer 2
    k_gemm2<<<(N + 127) / 128, TB, 0, stream>>>(H1, W2, T2, N);
    k_scatter2<<<(E + 7) / 8, TB, 0, stream>>>(ei, dinv, T2, H2, E);

    // log_softmax output
    k_final<<<(N + 7) / 8, TB, 0, stream>>>(H2, T2, dinv, b2, (float*)d_out, N);
}